// GINLayer_48756468744913
// MI455X (gfx1250) — compile-verified
//
#include <hip/hip_runtime.h>

typedef __attribute__((ext_vector_type(2))) float v2f;
typedef __attribute__((ext_vector_type(8))) float v8f;

#define B_  2
#define N_  50000
#define D_  64
#define E_  800000
#define EPS 0.0f

// ---------------------------------------------------------------------------
// Kernel 1: hpre = (1+EPS) * x   (float4 vectorized; 6.4M floats)
// ---------------------------------------------------------------------------
__global__ void gin_init(const float* __restrict__ x, float* __restrict__ hpre, int n4) {
    int i = blockIdx.x * blockDim.x + threadIdx.x;
    if (i < n4) {
        float4 v = ((const float4*)x)[i];
        const float s = 1.0f + EPS;
        v.x *= s; v.y *= s; v.z *= s; v.w *= s;
        ((float4*)hpre)[i] = v;
    }
}

// ---------------------------------------------------------------------------
// Kernel 2: scatter-add  hpre[b, dst[e], d] += x[b, src[e], d]
// One thread per (edge, dim); loops over batch. Lanes cover consecutive d ->
// coalesced gathers and coalesced f32 atomics (global_atomic_add_f32).
// edge_index is int64, shape (2,E): rows [src..., dst...]
// ---------------------------------------------------------------------------
__global__ void gin_scatter(const float* __restrict__ x,
                            const long long* __restrict__ ei,
                            float* __restrict__ hpre) {
    long long gid = (long long)blockIdx.x * blockDim.x + threadIdx.x;
    int e = (int)(gid >> 6);
    int d = (int)(gid & 63);
    if (e >= E_) return;
    int s = (int)ei[e];
    int t = (int)ei[(long long)E_ + e];
#pragma unroll
    for (int b = 0; b < B_; ++b) {
        float v = x[((long long)b * N_ + s) * D_ + d];
        unsafeAtomicAdd(&hpre[((long long)b * N_ + t) * D_ + d], v);
    }
}

// ---------------------------------------------------------------------------
// Kernel 3: Y[M,64] = act(X[M,64] @ W[64,64] + bias)    (fp32 WMMA 16x16x4)
//
// One wave computes a 16(M) x 64(N) strip: 4 accumulator tiles, 16 K-steps.
// W is staged transposed in LDS (sWt[col*64+row]) so each B-fragment
// (rows k..k+3 of one 16-col tile) is a single ds_load_b64 per lane.
//
// A-frag (16x4 f32): lane l<16 holds {X[m0+l, k], X[m0+l, k+1]},
//                    lane l>=16 holds {X[m0+l-16, k+2], X[m0+l-16, k+3]}.
// B-frag (4x16 f32): lanes 0-15 rows {k,k+1}, lanes 16-31 rows {k+2,k+3}.
// C/D (16x16 f32, 8 VGPRs): vgpr i -> row m0 + 8*half + i, col 16*t + (l&15).
//
// In-place safe: a wave reads only its own 16 rows (all consumed by the WMMA
// chain) before the epilogue stores those same rows.
// ---------------------------------------------------------------------------
template <bool RELU>
__global__ void gin_gemm(const float* __restrict__ X, const float* __restrict__ W,
                         const float* __restrict__ bias, float* __restrict__ Y, int M) {
    __shared__ float sWt[64 * 64];  // 16 KB, W transposed: sWt[col*64 + row]

    for (int idx = threadIdx.x; idx < 64 * 64; idx += blockDim.x) {
        int r = idx >> 6, c = idx & 63;
        sWt[c * 64 + r] = W[idx];
    }
    __syncthreads();

    const int wave = threadIdx.x >> 5;
    const int lane = threadIdx.x & 31;
    const int m0 = (blockIdx.x * (blockDim.x >> 5) + wave) * 16;
    if (m0 >= M) return;  // wave-uniform: EXEC stays all-ones for WMMA

    const int half = lane >> 4;   // 0: lanes 0-15, 1: lanes 16-31
    const int l16  = lane & 15;

    v8f acc[4] = {};  // 4 N-tiles of 16 cols each

    const float* xrow = X + (long long)(m0 + l16) * 64 + half * 2;

#pragma unroll
    for (int k = 0; k < 64; k += 4) {
        v2f a = *(const v2f*)(xrow + k);          // global_load_b64
        const int krow = k + half * 2;
#pragma unroll
        for (int t = 0; t < 4; ++t) {
            v2f b = *(const v2f*)&sWt[(t * 16 + l16) * 64 + krow];  // ds_load_b64
            acc[t] = __builtin_amdgcn_wmma_f32_16x16x4_f32(
                false, a, false, b, (short)0, acc[t], false, false);
        }
    }

    const int mlo = m0 + half * 8;
#pragma unroll
    for (int t = 0; t < 4; ++t) {
        const int col = t * 16 + l16;
        const float bv = bias[col];
#pragma unroll
        for (int i = 0; i < 8; ++i) {
            float v = acc[t][i] + bv;
            if (RELU) v = fmaxf(v, 0.0f);
            Y[(long long)(mlo + i) * 64 + col] = v;
        }
    }
}

// ---------------------------------------------------------------------------
extern "C" void kernel_launch(void* const* d_in, const int* in_sizes, int n_in,
                              void* d_out, int out_size, void* d_ws, size_t ws_size,
                              hipStream_t stream) {
    const float*     x  = (const float*)d_in[0];
    const long long* ei = (const long long*)d_in[1];
    const float*     W1 = (const float*)d_in[2];
    const float*     b1 = (const float*)d_in[3];
    const float*     W2 = (const float*)d_in[4];
    const float*     b2 = (const float*)d_in[5];
    float* out  = (float*)d_out;
    float* hpre = (float*)d_ws;  // B*N*64 floats = 25.6 MB scratch

    // hpre = (1+eps)*x
    const int n4 = (B_ * N_ * D_) / 4;
    gin_init<<<(n4 + 255) / 256, 256, 0, stream>>>(x, hpre, n4);

    // hpre += scatter(x over edges)
    const long long sc = (long long)E_ * 64;
    gin_scatter<<<(int)((sc + 255) / 256), 256, 0, stream>>>(x, ei, hpre);

    // h1 = relu(hpre @ W1 + b1) -> d_out ; out = h1 @ W2 + b2 (in place)
    const int M = B_ * N_;                 // 100000 (multiple of 16)
    const int grid = (M + 127) / 128;      // 8 waves/block * 16 rows/wave
    gin_gemm<true ><<<grid, 256, 0, stream>>>(hpre, W1, b1, out, M);
    gin_gemm<false><<<grid, 256, 0, stream>>>(out,  W2, b2, out, M);
}